// TreeRCNN_57578331570791
// MI455X (gfx1250) — compile-verified
//
#include <hip/hip_runtime.h>
#include <hip/hip_bf16.h>
#include <stdint.h>

// Problem constants (from reference)
#define NPTS     65536
#define NANCH    1024
#define MAXP     512
// Tiling
#define NTHREADS 256
#define TILE     2048                // points per TDM tile
#define NTILES   (NPTS / TILE)       // 32
#define TFLOATS  (TILE * 3)          // 6144 floats = 24 KB per buffer
#define APB      4                   // anchors per block (amortize point stream)
#define PPT      (TILE / NTHREADS)   // 8 contiguous points per thread (keeps order)

typedef unsigned int u32x4 __attribute__((ext_vector_type(4)));
typedef int          i32x8 __attribute__((ext_vector_type(8)));
typedef int          i32x4 __attribute__((ext_vector_type(4)));

// Issue one TDM tile load: points[tileIdx*TILE .. +TILE) -> LDS starting at byte offset ldsByteOff.
// D# per CDNA5 ISA: group0 = {count/lds_addr/global_addr/type},
// group1 = {data_size, tensor_dim0/1, tile_dim0/1, tensor_dim0_stride}.
__device__ __forceinline__ void tdm_issue_tile(const float* pts, int tileIdx,
                                               unsigned ldsByteOff) {
  unsigned long long ga =
      (unsigned long long)(uintptr_t)(pts + (size_t)tileIdx * TFLOATS);

  u32x4 g0;
  g0[0] = 1u;                                        // count=1, user descriptor
  g0[1] = ldsByteOff;                                // lds_addr [63:32]
  g0[2] = (unsigned)(ga & 0xFFFFFFFFull);            // global_addr lo
  g0[3] = ((unsigned)(ga >> 32) & 0x01FFFFFFu)       // global_addr[56:32]
          | (2u << 30);                              // type = 2 ("image")

  const unsigned td0 = NPTS * 3;                     // tensor_dim0 = 196608 f32
  const unsigned td1 = 1u;                           // tensor_dim1
  const unsigned tl0 = TFLOATS;                      // tile_dim0 = 6144 f32
  const unsigned tl1 = 1u;                           // tile_dim1
  const unsigned long long s0 = NPTS * 3;            // tensor_dim0_stride

  i32x8 g1;
  g1[0] = (int)(2u << 16);                           // wg_mask=0, data_size=2 (4B)
  g1[1] = (int)((td0 & 0xFFFFu) << 16);              // barrier_addr=0 | td0[15:0]
  g1[2] = (int)((td0 >> 16) | ((td1 & 0xFFFFu) << 16)); // td0[31:16] | td1[15:0]
  g1[3] = (int)((td1 >> 16) | (tl0 << 16));          // td1[31:16] | tile_dim0
  g1[4] = (int)(tl1);                                // tile_dim1=1, tile_dim2=0
  g1[5] = (int)(unsigned)(s0 & 0xFFFFFFFFull);       // stride0[31:0]
  g1[6] = (int)((unsigned)(s0 >> 32) & 0xFFFFu);     // stride0[47:32] | stride1 lo=0
  g1[7] = 0;

  i32x4 z4 = {0, 0, 0, 0};                           // groups 2/3 unused (<=2D)
  i32x8 z8 = {0, 0, 0, 0, 0, 0, 0, 0};               // extra group (clang-23 form)
  __builtin_amdgcn_tensor_load_to_lds(g0, g1, z4, z4, z8, 0);
}

__global__ __launch_bounds__(NTHREADS)
void tree_rcnn_group_points(const float* __restrict__ points,
                            const float* __restrict__ anchors,
                            float* __restrict__ out,
                            int* __restrict__ counts) {
  // NOTE: tileBuf is the first/largest shared object -> assumed at LDS offset 0
  // (TDM descriptors use raw LDS byte offsets 0 and TFLOATS*4).
  __shared__ float tileBuf[2][TFLOATS];              // 48 KB double buffer
  __shared__ int   wsum[APB][NTHREADS / 32];

  const int tid  = threadIdx.x;
  const int lane = tid & 31;                         // wave32
  const int wid  = tid >> 5;                         // 8 waves
  // Scalarized wave id: forces an s_cmp/s_cbranch (uniform branch) around the
  // TDM issue region. Tensor ops IGNORE EXEC, so an EXEC-masked lowering would
  // wrongly issue from all 8 waves — a scalar branch guarantees wave-0-only.
  const int swid = __builtin_amdgcn_readfirstlane(wid);
  const int a0   = blockIdx.x * APB;

  // Anchor box params (broadcast loads).
  float cx[APB], cy[APB], xlo[APB], xhi[APB], ylo[APB], yhi[APB], hh[APB];
#pragma unroll
  for (int aa = 0; aa < APB; ++aa) {
    const float* A = anchors + (size_t)(a0 + aa) * 6;
    const float c0 = A[0], c1 = A[1], w = A[3], l = A[4], h = A[5];
    cx[aa] = c0; cy[aa] = c1;
    xlo[aa] = c0 - 0.5f * w; xhi[aa] = c0 + 0.5f * w;
    ylo[aa] = c1 - 0.5f * l; yhi[aa] = c1 + 0.5f * l;
    hh[aa] = h;
  }

  // Zero-pad our output slab (harness poisons d_out).
  for (int i = tid; i < APB * MAXP * 3; i += NTHREADS)
    out[(size_t)a0 * MAXP * 3 + i] = 0.0f;

  int written[APB];
#pragma unroll
  for (int aa = 0; aa < APB; ++aa) written[aa] = 0;

  // Prologue: wave 0 kicks off tile 0 into buffer 0.
  if (swid == 0) tdm_issue_tile(points, 0, 0u);

  for (int t = 0; t < NTILES; ++t) {
    // Wave 0: prefetch tile t+1 into the other buffer, then wait for tile t.
    if (swid == 0) {
      if (t + 1 < NTILES) {
        tdm_issue_tile(points, t + 1,
                       ((t + 1) & 1) ? (unsigned)(TFLOATS * 4) : 0u);
        __builtin_amdgcn_s_wait_tensorcnt((short)1);  // tile t done, t+1 in flight
      } else {
        __builtin_amdgcn_s_wait_tensorcnt((short)0);
      }
    }
    __syncthreads();                                  // tile t visible to all waves

    // Each thread owns PPT *contiguous* points -> compaction preserves index order.
    const float* tb = &tileBuf[t & 1][0];
    float px[PPT], py[PPT], pz[PPT];
#pragma unroll
    for (int j = 0; j < PPT; ++j) {
      const int p = tid * PPT + j;
      px[j] = tb[p * 3 + 0];
      py[j] = tb[p * 3 + 1];
      pz[j] = tb[p * 3 + 2];
    }

    unsigned msk[APB]; int cnt[APB]; int incl[APB];
#pragma unroll
    for (int aa = 0; aa < APB; ++aa) {
      unsigned m = 0; int c = 0;
#pragma unroll
      for (int j = 0; j < PPT; ++j) {
        const bool in = (px[j] >= xlo[aa]) & (px[j] <= xhi[aa]) &
                        (py[j] >= ylo[aa]) & (py[j] <= yhi[aa]) &
                        (pz[j] >= 0.0f)    & (pz[j] <= hh[aa]);
        m |= (in ? 1u : 0u) << j;
        c += in ? 1 : 0;
      }
      msk[aa] = m; cnt[aa] = c;
      // Wave-level inclusive scan (wave32).
      int v = c;
#pragma unroll
      for (int d = 1; d < 32; d <<= 1) {
        const int o = __shfl_up(v, d, 32);
        if (lane >= d) v += o;
      }
      incl[aa] = v;
      if (lane == 31) wsum[aa][wid] = v;              // wave totals
    }
    __syncthreads();

#pragma unroll
    for (int aa = 0; aa < APB; ++aa) {
      int waveOff = 0, tileTotal = 0;
#pragma unroll
      for (int w = 0; w < NTHREADS / 32; ++w) {
        const int s = wsum[aa][w];
        tileTotal += s;
        if (w < wid) waveOff += s;
      }
      const int base = written[aa] + waveOff + (incl[aa] - cnt[aa]);
      const unsigned m = msk[aa];
      float* dstA = out + (size_t)(a0 + aa) * MAXP * 3;
      int k = 0;
#pragma unroll
      for (int j = 0; j < PPT; ++j) {
        if (m & (1u << j)) {
          const int o = base + k;
          if (o < MAXP) {
            dstA[o * 3 + 0] = px[j] - cx[aa];
            dstA[o * 3 + 1] = py[j] - cy[aa];
            dstA[o * 3 + 2] = pz[j];                  // center z is 0
          }
          ++k;
        }
      }
      written[aa] += tileTotal;                       // identical in all threads
    }
    __syncthreads();                                  // protect wsum + buffer reuse
  }

  if (tid == 0) {
#pragma unroll
    for (int aa = 0; aa < APB; ++aa) counts[a0 + aa] = written[aa];
  }
}

extern "C" void kernel_launch(void* const* d_in, const int* in_sizes, int n_in,
                              void* d_out, int out_size, void* d_ws, size_t ws_size,
                              hipStream_t stream) {
  (void)in_sizes; (void)n_in; (void)out_size; (void)d_ws; (void)ws_size;
  const float* points  = (const float*)d_in[0];   // (65536,3) f32
  const float* anchors = (const float*)d_in[1];   // (1024,6)  f32
  float* out   = (float*)d_out;                               // (1024,512,3) f32
  int*   cnts  = (int*)((float*)d_out + (size_t)NANCH * MAXP * 3); // (1024,) i32
  tree_rcnn_group_points<<<dim3(NANCH / APB), dim3(NTHREADS), 0, stream>>>(
      points, anchors, out, cnts);
}